// EquivariantMPLayer_41317585387560
// MI455X (gfx1250) — compile-verified
//
#include <hip/hip_runtime.h>

typedef float v2f __attribute__((ext_vector_type(2)));
typedef float v8f __attribute__((ext_vector_type(8)));

constexpr int Nn = 40000;   // nodes
constexpr int Ee = 640000;  // edges
constexpr int Cc = 128;     // in channels
constexpr int Hh = 256;     // hidden
constexpr int Rr = 16;      // rank
constexpr int Tt = 4;       // edge types
constexpr int Mm = 257;     // message size

constexpr size_t alignup(size_t x) { return (x + 255) & ~(size_t)255; }
// workspace layout (bytes)
constexpr size_t WP_OFF   = 0;                                             // packed W_t  [T][128 k2][256 n][2]
constexpr size_t WLAST_OFF= alignup(WP_OFF   + (size_t)Tt*128*256*2*4);    // W_t column k=256  [T][256]
constexpr size_t MP_OFF   = alignup(WLAST_OFF+ (size_t)Tt*Hh*4);           // packed mlp_w^T [192 k2][256][2]
constexpr size_t RP_OFF   = alignup(MP_OFF   + (size_t)192*256*2*4);       // packed W_res^T [64 k2][256][2]
constexpr size_t TPTR_OFF = alignup(RP_OFF   + (size_t)64*256*2*4);        // type_ptr[5]
constexpr size_t TCUR_OFF = alignup(TPTR_OFF + 5*4);                       // type counts/cursors[4]
constexpr size_t CPTR_OFF = alignup(TCUR_OFF + 4*4);                       // col_ptr[N+1]
constexpr size_t CCUR_OFF = alignup(CPTR_OFF + (size_t)(Nn+1)*4);          // col counts/cursors[N]
constexpr size_t TORD_OFF = alignup(CCUR_OFF + (size_t)Nn*4);              // edges sorted by type [E]
constexpr size_t CORD_OFF = alignup(TORD_OFF + (size_t)Ee*4);              // edges sorted by col  [E]
constexpr size_t AGGR_OFF = alignup(CORD_OFF + (size_t)Ee*4);              // aggr [N][H]
constexpr size_t MBUF_OFF = alignup(AGGR_OFF + (size_t)Nn*Hh*4);           // m    [E][H]

__device__ __forceinline__ v8f wmma_f32(v2f a, v2f b, v8f c) {
  // V_WMMA_F32_16X16X4_F32: D(16x16) = A(16x4) * B(4x16) + C
  return __builtin_amdgcn_wmma_f32_16x16x4_f32(false, a, false, b, (short)0, c,
                                               false, false);
}

// One 32-k chunk of a 16xH GEMM: A rows from abase (per-lane), B from LDS chunk Wc.
// 64 B-pairs consumed through a 4-slot rotating buffer so 4 ds_loads stay in flight.
__device__ __forceinline__ void gemm_chunk(const float* abase, const float* Wc,
                                           int hi, int lo, v8f* acc) {
  v2f av[8];
#pragma unroll
  for (int s = 0; s < 8; ++s) av[s] = *(const v2f*)(abase + s * 4 + hi * 2);
  v2f bq[4][2];
#pragma unroll
  for (int P = 0; P < 4; ++P) {
    int s = P >> 3, q = P & 7;
    const float* p0 = &Wc[((s * 2 + hi) * 256 + (2 * q) * 16 + lo) * 2];
    bq[P & 3][0] = *(const v2f*)p0;
    bq[P & 3][1] = *(const v2f*)(p0 + 32);
  }
#pragma unroll
  for (int P = 0; P < 64; ++P) {
    int s = P >> 3, q = P & 7;
    acc[2 * q]     = wmma_f32(av[s], bq[P & 3][0], acc[2 * q]);
    acc[2 * q + 1] = wmma_f32(av[s], bq[P & 3][1], acc[2 * q + 1]);
    if (P + 4 < 64) {
      int s2 = (P + 4) >> 3, q2 = (P + 4) & 7;
      const float* p0 = &Wc[((s2 * 2 + hi) * 256 + (2 * q2) * 16 + lo) * 2];
      bq[(P + 4) & 3][0] = *(const v2f*)p0;
      bq[(P + 4) & 3][1] = *(const v2f*)(p0 + 32);
    }
  }
}

// ---------------- setup kernels ----------------

__global__ void zero_counts(int* __restrict__ tcur, int* __restrict__ ccur) {
  int i = blockIdx.x * 256 + threadIdx.x;
  if (i < 4) tcur[i] = 0;
  if (i < Nn) ccur[i] = 0;
}

// W_t[n][k] = W_sh[n][k] + sum_r A_w[t][n][r]*B_w[t][r][k], packed pairwise over k
__global__ void pack_wt(const float* __restrict__ Wsh, const float* __restrict__ Aw,
                        const float* __restrict__ Bw, float* __restrict__ Wp,
                        float* __restrict__ Wlast) {
  int b = blockIdx.x;       // T*257 blocks
  int t = b / Mm, k = b % Mm;
  __shared__ float bw[Rr];
  if (threadIdx.x < Rr) bw[threadIdx.x] = Bw[((size_t)t * Rr + threadIdx.x) * Mm + k];
  __syncthreads();
  int n = threadIdx.x;      // 256 threads
  float v = Wsh[(size_t)n * Mm + k];
  const float* aw = Aw + ((size_t)t * Hh + n) * Rr;
#pragma unroll
  for (int r = 0; r < Rr; ++r) v += aw[r] * bw[r];
  if (k < 256)
    Wp[(((size_t)t * 128 + (k >> 1)) * 256 + n) * 2 + (k & 1)] = v;
  else
    Wlast[t * Hh + n] = v;
}

__global__ void pack_node(const float* __restrict__ mlpw, const float* __restrict__ Wres,
                          float* __restrict__ Mp, float* __restrict__ Rp) {
  int idx = blockIdx.x * 256 + threadIdx.x;  // 512 blocks
  constexpr int MPN = 192 * 256 * 2;
  if (idx < MPN) {
    int c = idx & 1, n = (idx >> 1) & 255, k2 = idx >> 9;
    Mp[idx] = mlpw[(size_t)n * (Cc + Hh) + k2 * 2 + c];
  } else {
    int j = idx - MPN;
    if (j < 64 * 256 * 2) {
      int c = j & 1, n = (j >> 1) & 255, k2 = j >> 9;
      Rp[j] = Wres[(size_t)n * Cc + k2 * 2 + c];
    }
  }
}

__global__ void hist_edges(const float* __restrict__ eattr, const int* __restrict__ eidx,
                           int* __restrict__ tcnt, int* __restrict__ ccnt) {
  int e = blockIdx.x * 256 + threadIdx.x;
  if (e >= Ee) return;
  int t = (int)eattr[(size_t)e * 2];
  int c = eidx[Ee + e];
  atomicAdd(&tcnt[t], 1);
  atomicAdd(&ccnt[c], 1);
}

__global__ void scan_type(int* __restrict__ tcur, int* __restrict__ tptr) {
  if (threadIdx.x == 0 && blockIdx.x == 0) {
    int run = 0;
    for (int t = 0; t < Tt; ++t) { int c = tcur[t]; tptr[t] = run; tcur[t] = run; run += c; }
    tptr[Tt] = run;
  }
}

__global__ void scan_col(int* __restrict__ ccur, int* __restrict__ cptr) {
  __shared__ int part[256];
  int t = threadIdx.x;
  int per = (Nn + 255) / 256;
  int b = t * per, e = b + per; if (e > Nn) e = Nn;
  int s = 0;
  for (int i = b; i < e; ++i) s += ccur[i];
  part[t] = s;
  __syncthreads();
  if (t == 0) {
    int run = 0;
    for (int i = 0; i < 256; ++i) { int x = part[i]; part[i] = run; run += x; }
    cptr[Nn] = run;
  }
  __syncthreads();
  int run = part[t];
  for (int i = b; i < e; ++i) { int x = ccur[i]; cptr[i] = run; ccur[i] = run; run += x; }
}

__global__ void scatter_edges(const float* __restrict__ eattr, const int* __restrict__ eidx,
                              int* __restrict__ tcur, int* __restrict__ ccur,
                              int* __restrict__ tord, int* __restrict__ cord) {
  int e = blockIdx.x * 256 + threadIdx.x;
  if (e >= Ee) return;
  int t = (int)eattr[(size_t)e * 2];
  int c = eidx[Ee + e];
  int p = atomicAdd(&tcur[t], 1); tord[p] = e;
  int q = atomicAdd(&ccur[c], 1); cord[q] = e;
}

// ---------------- main edge GEMM: m[e] = W_t @ msg[e] + A_b[t] ----------------
// 8 waves/block, each wave one 16-edge tile (type-uniform), full H=256 in registers.
__global__ __launch_bounds__(256) void edge_gemm(
    const float* __restrict__ embed, const float* __restrict__ eattr,
    const int* __restrict__ eidx, const int* __restrict__ tord,
    const int* __restrict__ tptr, const float* __restrict__ Wp,
    const float* __restrict__ Wlast, const float* __restrict__ Ab,
    float* __restrict__ mbuf) {
  __shared__ __align__(16) float Wc[16 * 256 * 2];  // 32KB W chunk (32 k-values)
  __shared__ int dS[8 * 16], dT[8 * 16], dE[8 * 16];
  __shared__ float dD[8 * 16];

  const int tid = threadIdx.x;
  const int wave = tid >> 5, lane = tid & 31;
  const int lo = lane & 15, hi = lane >> 4;

  int tp[Tt + 1];
#pragma unroll
  for (int i = 0; i <= Tt; ++i) tp[i] = tptr[i];

  // map block -> (type, block index within type); whole block is one type
  int myType = -1, blkInType = 0, wgAcc = 0;
#pragma unroll
  for (int t = 0; t < Tt; ++t) {
    int cnt = tp[t + 1] - tp[t];
    int tiles = (cnt + 15) >> 4;
    int wgs = (tiles + 7) >> 3;
    if (myType < 0 && (int)blockIdx.x < wgAcc + wgs) { myType = t; blkInType = (int)blockIdx.x - wgAcc; }
    wgAcc += wgs;
  }
  if (myType < 0) return;  // overshoot block: whole block exits before barriers
  const int t = myType;
  const int cnt = tp[t + 1] - tp[t];
  const int tiles = (cnt + 15) >> 4;
  const int tileIdx = blkInType * 8 + wave;
  const bool waveActive = tileIdx < tiles;

  if (waveActive && hi == 0) {
    int p = tp[t] + tileIdx * 16 + lo;
    int e = (p < tp[t + 1]) ? tord[p] : -1;
    int s = 0, g = 0; float dd = 0.f;
    if (e >= 0) { s = eidx[e]; g = eidx[Ee + e]; dd = eattr[(size_t)e * 2 + 1]; }
    dS[wave * 16 + lo] = s; dT[wave * 16 + lo] = g;
    dE[wave * 16 + lo] = e; dD[wave * 16 + lo] = dd;
  }
  __syncthreads();

  const float* aSrc = embed;
  const float* aTgt = embed;
  if (waveActive) {
    aSrc = embed + (size_t)dS[wave * 16 + lo] * Cc;
    aTgt = embed + (size_t)dT[wave * 16 + lo] * Cc;
  }

  v8f acc[16];
#pragma unroll
  for (int nt = 0; nt < 16; ++nt) acc[nt] = (v8f)0.f;

  const float* gw = Wp + (size_t)t * 128 * 512;  // [k2][n][2] for this type
  // chunks 0..3 read the src half of msg (k<128); chunks 4..7 the tgt half.
#pragma unroll 1
  for (int chunk = 0; chunk < 8; ++chunk) {      // 32 k-values per chunk
    const float* src = gw + (size_t)chunk * 16 * 512;
    for (int i = tid * 4; i < 16 * 256 * 2; i += 256 * 4)
      *(float4*)&Wc[i] = *(const float4*)&src[i];
    __syncthreads();
    if (waveActive) {
      const float* abase = (chunk < 4) ? (aSrc + chunk * 32) : (aTgt + (chunk - 4) * 32);
      gemm_chunk(abase, Wc, hi, lo, acc);
    }
    __syncthreads();
  }

  if (waveActive) {
    float dd[8]; int ee[8];
#pragma unroll
    for (int v = 0; v < 8; ++v) {
      int ml = v + 8 * hi;
      dd[v] = dD[wave * 16 + ml];
      ee[v] = dE[wave * 16 + ml];
    }
#pragma unroll
    for (int nt = 0; nt < 16; ++nt) {
      int n = nt * 16 + lo;
      float wl = Wlast[t * Hh + n];   // dist column (k = 256)
      float bias = Ab[t * Hh + n];
#pragma unroll
      for (int v = 0; v < 8; ++v) {
        float val = acc[nt][v] + dd[v] * wl + bias;
        if (ee[v] >= 0) mbuf[(size_t)ee[v] * Hh + n] = val;
      }
    }
  }
}

// ---------------- segment mean (deterministic: sort ids first) ----------------
__global__ __launch_bounds__(256) void agg_mean(const float* __restrict__ mbuf,
                                                const int* __restrict__ cptr,
                                                const int* __restrict__ cord,
                                                float* __restrict__ aggr) {
  __shared__ int ids[1024];
  int n = blockIdx.x;
  int beg = cptr[n], end = cptr[n + 1];
  int degFull = end - beg;
  int deg = degFull > 1024 ? 1024 : degFull;
  for (int i = threadIdx.x; i < deg; i += 256) ids[i] = cord[beg + i];
  __syncthreads();
  // odd-even transposition sort (deg is small, avg 16)
  for (int it = 0; it < deg; ++it) {
    int ph = it & 1;
    for (int j = threadIdx.x;; j += 256) {
      int idx = 2 * j + ph;
      if (idx + 1 >= deg) break;
      int a = ids[idx], b = ids[idx + 1];
      if (a > b) { ids[idx] = b; ids[idx + 1] = a; }
    }
    __syncthreads();
  }
  int h = threadIdx.x;  // H == blockDim
  float s = 0.f;
  for (int i = 0; i < deg; ++i) s += mbuf[(size_t)ids[i] * Hh + h];
  float denom = (float)(degFull > 0 ? degFull : 1);
  aggr[(size_t)n * Hh + h] = s / denom;
}

// ---------------- node update: relu(hcat@mlp^T + b) + embed@Wres^T ----------------
__global__ __launch_bounds__(256) void node_gemm(
    const float* __restrict__ embed, const float* __restrict__ aggr,
    const float* __restrict__ Mp, const float* __restrict__ Rp,
    const float* __restrict__ mlpb, float* __restrict__ out) {
  __shared__ __align__(16) float Wc[16 * 256 * 2];
  const int tid = threadIdx.x;
  const int wave = tid >> 5, lane = tid & 31;
  const int lo = lane & 15, hi = lane >> 4;

  const int tiles = Nn / 16;  // 2500
  const int tile = (int)blockIdx.x * 8 + wave;
  const bool waveActive = tile < tiles;
  const int nodeBase = waveActive ? tile * 16 : 0;
  const float* aE = embed + (size_t)(nodeBase + lo) * Cc;
  const float* aG = aggr + (size_t)(nodeBase + lo) * Hh;

  v8f acc[16];
#pragma unroll
  for (int nt = 0; nt < 16; ++nt) acc[nt] = (v8f)0.f;

  // phase 1: mlp over K = C+H = 384 (12 chunks of 32); chunks 0..3 embed, 4..11 aggr
#pragma unroll 1
  for (int chunk = 0; chunk < 12; ++chunk) {
    const float* src = Mp + (size_t)chunk * 16 * 512;
    for (int i = tid * 4; i < 16 * 256 * 2; i += 256 * 4)
      *(float4*)&Wc[i] = *(const float4*)&src[i];
    __syncthreads();
    if (waveActive) {
      const float* abase = (chunk < 4) ? (aE + chunk * 32) : (aG + (chunk - 4) * 32);
      gemm_chunk(abase, Wc, hi, lo, acc);
    }
    __syncthreads();
  }
  // bias + relu in-register
  if (waveActive) {
#pragma unroll
    for (int nt = 0; nt < 16; ++nt) {
      float b = mlpb[nt * 16 + lo];
#pragma unroll
      for (int v = 0; v < 8; ++v) acc[nt][v] = fmaxf(acc[nt][v] + b, 0.f);
    }
  }
  // phase 2: residual embed @ Wres^T over K = 128 (4 chunks), accumulate on top
#pragma unroll 1
  for (int chunk = 0; chunk < 4; ++chunk) {
    const float* src = Rp + (size_t)chunk * 16 * 512;
    for (int i = tid * 4; i < 16 * 256 * 2; i += 256 * 4)
      *(float4*)&Wc[i] = *(const float4*)&src[i];
    __syncthreads();
    if (waveActive) gemm_chunk(aE + chunk * 32, Wc, hi, lo, acc);
    __syncthreads();
  }
  if (waveActive) {
#pragma unroll
    for (int nt = 0; nt < 16; ++nt) {
      int n = nt * 16 + lo;
#pragma unroll
      for (int v = 0; v < 8; ++v)
        out[(size_t)(nodeBase + v + 8 * hi) * Hh + n] = acc[nt][v];
    }
  }
}

extern "C" void kernel_launch(void* const* d_in, const int* in_sizes, int n_in,
                              void* d_out, int out_size, void* d_ws, size_t ws_size,
                              hipStream_t stream) {
  const float* embed = (const float*)d_in[0];
  const float* eattr = (const float*)d_in[2];
  const float* Wres  = (const float*)d_in[3];
  const float* Aw    = (const float*)d_in[4];
  const float* Ab    = (const float*)d_in[5];
  const float* Bw    = (const float*)d_in[6];
  const float* Wsh   = (const float*)d_in[7];
  const float* mlpw  = (const float*)d_in[8];
  const float* mlpb  = (const float*)d_in[9];
  const int*   eidx  = (const int*)d_in[10];
  float* out = (float*)d_out;

  char* ws = (char*)d_ws;
  float* Wp    = (float*)(ws + WP_OFF);
  float* Wlast = (float*)(ws + WLAST_OFF);
  float* Mp    = (float*)(ws + MP_OFF);
  float* Rp    = (float*)(ws + RP_OFF);
  int*   tptr  = (int*)(ws + TPTR_OFF);
  int*   tcur  = (int*)(ws + TCUR_OFF);
  int*   cptr  = (int*)(ws + CPTR_OFF);
  int*   ccur  = (int*)(ws + CCUR_OFF);
  int*   tord  = (int*)(ws + TORD_OFF);
  int*   cord  = (int*)(ws + CORD_OFF);
  float* aggr  = (float*)(ws + AGGR_OFF);
  float* mbuf  = (float*)(ws + MBUF_OFF);

  zero_counts<<<(Nn + 255) / 256, 256, 0, stream>>>(tcur, ccur);
  pack_wt<<<Tt * Mm, 256, 0, stream>>>(Wsh, Aw, Bw, Wp, Wlast);
  pack_node<<<512, 256, 0, stream>>>(mlpw, Wres, Mp, Rp);
  hist_edges<<<(Ee + 255) / 256, 256, 0, stream>>>(eattr, eidx, tcur, ccur);
  scan_type<<<1, 32, 0, stream>>>(tcur, tptr);
  scan_col<<<1, 256, 0, stream>>>(ccur, cptr);
  scatter_edges<<<(Ee + 255) / 256, 256, 0, stream>>>(eattr, eidx, tcur, ccur, tord, cord);
  // worst case: sum_t ceil(ceil(cnt_t/16)/8) <= E/128 + 2*T
  edge_gemm<<<Ee / 128 + 2 * Tt, 256, 0, stream>>>(embed, eattr, eidx, tord, tptr,
                                                   Wp, Wlast, Ab, mbuf);
  agg_mean<<<Nn, 256, 0, stream>>>(mbuf, cptr, cord, aggr);
  node_gemm<<<(Nn / 16 + 7) / 8, 256, 0, stream>>>(embed, aggr, Mp, Rp, mlpb, out);
}